// RGSE_5566277615911
// MI455X (gfx1250) — compile-verified
//
#include <hip/hip_runtime.h>
#include <hip/hip_bf16.h>

// ---------- types ----------
typedef __attribute__((ext_vector_type(4))) float f4;
typedef __attribute__((ext_vector_type(2))) float v2f;
typedef __attribute__((ext_vector_type(8))) float v8f;

// Problem constants (fixed by the reference)
#define B_    16
#define C_    256
#define H_    224
#define W_    224
#define KP    7
#define HP    32            // H/K
#define WP    32            // W/K
#define P_    1024          // HP*WP
#define R_    16            // C/16
#define PLANE (H_*W_)       // 50176 floats per (b,c) plane
#define PLANE4 (PLANE/4)    // 12544 float4
#define ITERS  (PLANE4/256) // 49 float4 per thread @ 256 threads

// =====================================================================
// Kernel 1: fused avgpool(7x7) + per-channel Linear(P->1) reduction.
// One block per (b,c) plane. Weighted sum over the whole plane:
//   fused[b,c] = (sum_{h,w} x * w[c, (h/7)*32 + (w/7)]) / 49 + fusion_b[c]
// =====================================================================
__global__ __launch_bounds__(256)
void rgse_pool_fuse(const float* __restrict__ x,
                    const float* __restrict__ fusion_w,
                    const float* __restrict__ fusion_b,
                    float* __restrict__ fused) {
    __shared__ float wlds[P_];
    __shared__ float red[256];

    const unsigned bc = blockIdx.x;       // b*C + c
    const unsigned c  = bc & (C_ - 1);
    const unsigned t  = threadIdx.x;

    // stage fusion_w[c, :] (1024 floats) into LDS
    #pragma unroll
    for (int i = 0; i < P_ / 256; ++i)
        wlds[t + 256u * i] = fusion_w[(size_t)c * P_ + t + 256u * i];
    __syncthreads();

    const f4* __restrict__ xp = (const f4*)(x + (size_t)bc * PLANE);
    float acc = 0.0f;

    for (unsigned it = 0; it < ITERS; ++it) {
        unsigned i4 = t + it * 256u;      // float4 index within plane
        f4 v = xp[i4];
        unsigned row  = i4 / 56u;         // 56 float4 per 224-wide row
        unsigned col0 = (i4 - row * 56u) * 4u;
        unsigned base = (row / 7u) * WP;  // pooled-row offset
        acc += v.x * wlds[base + (col0       ) / 7u];
        acc += v.y * wlds[base + (col0 + 1u) / 7u];
        acc += v.z * wlds[base + (col0 + 2u) / 7u];
        acc += v.w * wlds[base + (col0 + 3u) / 7u];
    }

    red[t] = acc;
    __syncthreads();
    #pragma unroll
    for (unsigned s = 128; s > 0; s >>= 1) {
        if (t < s) red[t] += red[t + s];
        __syncthreads();
    }
    if (t == 0)
        fused[bc] = red[0] * (1.0f / 49.0f) + fusion_b[c];
}

// =====================================================================
// Kernel 2: SE FC on one wave32 with V_WMMA_F32_16X16X4_F32.
//   h    = relu(fused[16,256] @ fc1_w^T[256,16] + fc1_b)   (64 WMMA, K=256)
//   gate = h[16,16] @ fc2_w^T[16,256] + fc2_b              (16 N-tiles x 4 WMMA)
// A-layout (32-bit, 16x4): lane L holds M = L&15; VGPR0/1 hold K = koff/koff+1
// with koff = 2*(L>>4). B symmetric (lane = N). D: VGPR j = row j + 8*(L>>4),
// col = L&15. Bias (per-column) is preloaded into the accumulator.
// =====================================================================
__global__ __launch_bounds__(32)
void rgse_se_fc(const float* __restrict__ fused,
                const float* __restrict__ fc1_w, const float* __restrict__ fc1_b,
                const float* __restrict__ fc2_w, const float* __restrict__ fc2_b,
                float* __restrict__ gate) {
    __shared__ float hs[16 * 16];

    const int L    = threadIdx.x;     // 0..31, EXEC all ones
    const int mn   = L & 15;          // row (A) / col (B,D)
    const int koff = (L >> 4) * 2;    // K sub-offset for this half-wave
    const int mrow = 8 * (L >> 4);    // D-layout row base

    // ---- stage 1: h = relu(fused @ fc1_w^T + fc1_b) ----
    v8f acc;
    {
        float bv = fc1_b[mn];
        #pragma unroll
        for (int j = 0; j < 8; ++j) acc[j] = bv;
    }
    #pragma unroll 4
    for (int k0 = 0; k0 < C_; k0 += 4) {
        v2f a, b;
        a.x = fused[mn * C_ + k0 + koff];
        a.y = fused[mn * C_ + k0 + koff + 1];
        b.x = fc1_w[mn * C_ + k0 + koff];      // B[k][n] = fc1_w[n][k]
        b.y = fc1_w[mn * C_ + k0 + koff + 1];
        acc = __builtin_amdgcn_wmma_f32_16x16x4_f32(
            false, a, false, b, (short)0, acc, false, false);
    }
    #pragma unroll
    for (int j = 0; j < 8; ++j) acc[j] = fmaxf(acc[j], 0.0f);

    // D-layout -> row-major h[16][16] in LDS (transpose for reuse as A)
    #pragma unroll
    for (int j = 0; j < 8; ++j) hs[(mrow + j) * 16 + mn] = acc[j];
    __syncthreads();

    // ---- stage 2: gate = h @ fc2_w^T + fc2_b ----
    for (int n0 = 0; n0 < C_; n0 += 16) {
        v8f ct;
        {
            float bv = fc2_b[n0 + mn];
            #pragma unroll
            for (int j = 0; j < 8; ++j) ct[j] = bv;
        }
        #pragma unroll
        for (int k0 = 0; k0 < R_; k0 += 4) {
            v2f a, b;
            a.x = hs[mn * 16 + k0 + koff];
            a.y = hs[mn * 16 + k0 + koff + 1];
            b.x = fc2_w[(n0 + mn) * R_ + k0 + koff];   // B[k][n] = fc2_w[n][k]
            b.y = fc2_w[(n0 + mn) * R_ + k0 + koff + 1];
            ct = __builtin_amdgcn_wmma_f32_16x16x4_f32(
                false, a, false, b, (short)0, ct, false, false);
        }
        #pragma unroll
        for (int j = 0; j < 8; ++j)
            gate[(mrow + j) * C_ + n0 + mn] = ct[j];
    }
}

// =====================================================================
// Kernel 3: out = x * gate[b,c]. Pure streaming; non-temporal traffic.
// One block per (b,c) plane; gate load is block-uniform (scalar path).
// =====================================================================
__global__ __launch_bounds__(256)
void rgse_scale(const float* __restrict__ x,
                const float* __restrict__ gate,
                float* __restrict__ out) {
    const unsigned bc = blockIdx.x;
    const unsigned t  = threadIdx.x;
    const float g = gate[bc];

    const f4* __restrict__ xp = (const f4*)(x + (size_t)bc * PLANE);
    f4* __restrict__ op       = (f4*)(out + (size_t)bc * PLANE);

    #pragma unroll 7
    for (unsigned it = 0; it < ITERS; ++it) {
        unsigned i4 = t + it * 256u;
        f4 v = __builtin_nontemporal_load(&xp[i4]);
        v *= g;
        __builtin_nontemporal_store(v, &op[i4]);
    }
}

// =====================================================================
extern "C" void kernel_launch(void* const* d_in, const int* in_sizes, int n_in,
                              void* d_out, int out_size, void* d_ws, size_t ws_size,
                              hipStream_t stream) {
    const float* x        = (const float*)d_in[0];
    const float* fusion_w = (const float*)d_in[1];
    const float* fusion_b = (const float*)d_in[2];
    const float* fc1_w    = (const float*)d_in[3];
    const float* fc1_b    = (const float*)d_in[4];
    const float* fc2_w    = (const float*)d_in[5];
    const float* fc2_b    = (const float*)d_in[6];
    float* out = (float*)d_out;

    float* fused = (float*)d_ws;            // B*C = 4096 floats
    float* gate  = fused + B_ * C_;         // B*C = 4096 floats

    rgse_pool_fuse<<<B_ * C_, 256, 0, stream>>>(x, fusion_w, fusion_b, fused);
    rgse_se_fc<<<1, 32, 0, stream>>>(fused, fc1_w, fc1_b, fc2_w, fc2_b, gate);
    rgse_scale<<<B_ * C_, 256, 0, stream>>>(x, gate, out);
}